// MoERouter_84284438217167
// MI455X (gfx1250) — compile-verified
//
#include <hip/hip_runtime.h>
#include <hip/hip_bf16.h>
#include <math.h>

typedef __attribute__((ext_vector_type(2))) float v2f;
typedef __attribute__((ext_vector_type(4))) float v4f;
typedef __attribute__((ext_vector_type(8))) float v8f;

#define DMODEL 4096
#define NEXP   64
#define TOK_PER_WAVE 32
#define WAVES_PER_BLOCK 4
#define TOK_PER_BLOCK (TOK_PER_WAVE * WAVES_PER_BLOCK)   // 128
#define LDS_PITCH 68   // 64 + 4 pad: conflict-free epilogue, keeps 16B alignment

__global__ __launch_bounds__(128, 2)
void moe_router_wmma_f32(const float* __restrict__ x,     // (n_tokens, 4096)
                         const float* __restrict__ W,     // (64, 4096)
                         float* __restrict__ probs,       // (n_tokens, 64)
                         float* __restrict__ idx_out,     // (n_tokens, 2) as float values
                         float* __restrict__ w_out,       // (n_tokens, 2)
                         int n_tokens)
{
    __shared__ float lds[WAVES_PER_BLOCK][TOK_PER_WAVE][LDS_PITCH];

    const int lane = threadIdx.x & 31;
    const int wave = threadIdx.x >> 5;
    const int token_base = blockIdx.x * TOK_PER_BLOCK + wave * TOK_PER_WAVE;
    if (token_base >= n_tokens) return;   // wave-uniform; EXEC stays all-ones for WMMA

    // --- WMMA fragment addressing (ISA 16x4 f32 A / 4x16 f32 B layouts) ---
    const int m16   = lane & 15;           // A: token row within tile / B: expert col within tile
    const int khalf = (lane >> 4) * 2;     // lanes 16-31 carry K=2,3

    int arow0 = token_base + m16;
    int arow1 = token_base + 16 + m16;
    if (arow0 >= n_tokens) arow0 = n_tokens - 1;   // safe clamp (harmless dup work)
    if (arow1 >= n_tokens) arow1 = n_tokens - 1;

    const float* __restrict__ a0p = x + (size_t)arow0 * DMODEL + khalf;
    const float* __restrict__ a1p = x + (size_t)arow1 * DMODEL + khalf;
    const float* __restrict__ b0  = W + (size_t)(m16 +  0) * DMODEL + khalf;
    const float* __restrict__ b1  = W + (size_t)(m16 + 16) * DMODEL + khalf;
    const float* __restrict__ b2  = W + (size_t)(m16 + 32) * DMODEL + khalf;
    const float* __restrict__ b3  = W + (size_t)(m16 + 48) * DMODEL + khalf;

    v8f c0 = {}, c1 = {}, c2 = {}, c3 = {};   // tokens [0,16) x expert tiles 0..3
    v8f c4 = {}, c5 = {}, c6 = {}, c7 = {};   // tokens [16,32) x expert tiles 0..3

    // K loop: 1024 steps x 8 WMMAs. 6 b64 loads feed 8 WMMAs (ratio 0.75).
    // A rows hit HBM exactly once chip-wide; B addresses identical across all
    // waves -> WGP$/L2 broadcast, per-wave L2 B traffic halved vs 16-token tile.
    #pragma unroll 4
    for (int k = 0; k < DMODEL; k += 4) {
        v2f a0 = *(const v2f*)(a0p + k);
        v2f a1 = *(const v2f*)(a1p + k);
        v2f w0 = *(const v2f*)(b0 + k);
        v2f w1 = *(const v2f*)(b1 + k);
        v2f w2 = *(const v2f*)(b2 + k);
        v2f w3 = *(const v2f*)(b3 + k);
        c0 = __builtin_amdgcn_wmma_f32_16x16x4_f32(false, a0, false, w0, (short)0, c0, false, false);
        c1 = __builtin_amdgcn_wmma_f32_16x16x4_f32(false, a0, false, w1, (short)0, c1, false, false);
        c2 = __builtin_amdgcn_wmma_f32_16x16x4_f32(false, a0, false, w2, (short)0, c2, false, false);
        c3 = __builtin_amdgcn_wmma_f32_16x16x4_f32(false, a0, false, w3, (short)0, c3, false, false);
        c4 = __builtin_amdgcn_wmma_f32_16x16x4_f32(false, a1, false, w0, (short)0, c4, false, false);
        c5 = __builtin_amdgcn_wmma_f32_16x16x4_f32(false, a1, false, w1, (short)0, c5, false, false);
        c6 = __builtin_amdgcn_wmma_f32_16x16x4_f32(false, a1, false, w2, (short)0, c6, false, false);
        c7 = __builtin_amdgcn_wmma_f32_16x16x4_f32(false, a1, false, w3, (short)0, c7, false, false);
    }

    // --- Spill logits to LDS: c[t][v] -> token (lane>>4)*8+v (+16 for c4..c7),
    //     expert t*16+(lane&15). Lanes 0-15 take 16 consecutive banks, lanes
    //     16-31 take them +32 mod 64 -> conflict-free. ---
    {
        const int tokhi = (lane >> 4) << 3;
        const int e = lane & 15;
        #pragma unroll
        for (int v = 0; v < 8; ++v) {
            lds[wave][tokhi + v][e +  0] = c0[v];
            lds[wave][tokhi + v][e + 16] = c1[v];
            lds[wave][tokhi + v][e + 32] = c2[v];
            lds[wave][tokhi + v][e + 48] = c3[v];
            lds[wave][16 + tokhi + v][e +  0] = c4[v];
            lds[wave][16 + tokhi + v][e + 16] = c5[v];
            lds[wave][16 + tokhi + v][e + 32] = c6[v];
            lds[wave][16 + tokhi + v][e + 48] = c7[v];
        }
    }
    __syncthreads();

    // --- Softmax + top-2: every lane owns one token (pitch 68 -> banks rotate, no conflicts) ---
    {
        float* row = &lds[wave][lane][0];
        float mx = row[0];
        #pragma unroll
        for (int e = 1; e < NEXP; ++e) mx = fmaxf(mx, row[e]);

        float sum = 0.f, p1 = -1.f, p2 = -1.f;
        int i1 = 0, i2 = 0;
        #pragma unroll
        for (int e = 0; e < NEXP; ++e) {
            float p = __expf(row[e] - mx);   // unnormalized prob; order == logit order
            sum += p;
            row[e] = p;
            if (p > p1)      { p2 = p1; i2 = i1; p1 = p; i1 = e; }
            else if (p > p2) { p2 = p;  i2 = e; }
        }
        const float inv = 1.f / sum;
        #pragma unroll
        for (int e = 0; e < NEXP; ++e) row[e] *= inv;

        const int tg = token_base + lane;
        if (tg < n_tokens) {
            const float wnorm = 1.f / (p1 + p2);   // top-k renorm: softmax denom cancels
            idx_out[2 * tg + 0] = (float)i1;
            idx_out[2 * tg + 1] = (float)i2;
            w_out[2 * tg + 0] = p1 * wnorm;
            w_out[2 * tg + 1] = p2 * wnorm;
        }
    }
    __syncthreads();

    // --- Coalesced cooperative store of normalized probs: 32 tokens x 64 f32 per wave ---
    if (token_base + TOK_PER_WAVE <= n_tokens) {
        float* dst = probs + (size_t)token_base * NEXP;
        #pragma unroll
        for (int j = 0; j < 16; ++j) {
            const int f = j * 128 + lane * 4;       // flat f32 index in 32x64 tile
            const int t = f >> 6, e = f & 63;
            *(v4f*)(dst + f) = *(const v4f*)(&lds[wave][t][e]);
        }
    } else {
        for (int t = 0; t < TOK_PER_WAVE; ++t) {
            const int tg = token_base + t;
            if (tg < n_tokens) {
                probs[(size_t)tg * NEXP + 2 * lane + 0] = lds[wave][t][2 * lane + 0];
                probs[(size_t)tg * NEXP + 2 * lane + 1] = lds[wave][t][2 * lane + 1];
            }
        }
    }
}

extern "C" void kernel_launch(void* const* d_in, const int* in_sizes, int n_in,
                              void* d_out, int out_size, void* d_ws, size_t ws_size,
                              hipStream_t stream) {
    (void)n_in; (void)out_size; (void)d_ws; (void)ws_size;
    const float* x = (const float*)d_in[0];   // (4, 8192, 4096) f32
    const float* W = (const float*)d_in[1];   // (64, 4096) f32
    const int n_tokens = in_sizes[0] / DMODEL;  // 32768

    float* probs   = (float*)d_out;                               // n_tokens*64
    float* idx_out = probs + (size_t)n_tokens * NEXP;             // n_tokens*2 (indices as floats)
    float* w_out   = idx_out + (size_t)n_tokens * 2;              // n_tokens*2

    const int blocks = (n_tokens + TOK_PER_BLOCK - 1) / TOK_PER_BLOCK;  // 256
    moe_router_wmma_f32<<<blocks, 128, 0, stream>>>(x, W, probs, idx_out, w_out, n_tokens);
}